// MSDeformableAttention_90537910600288
// MI455X (gfx1250) — compile-verified
//
#include <hip/hip_runtime.h>

typedef __attribute__((ext_vector_type(2))) float v2f;
typedef __attribute__((ext_vector_type(8))) float v8f;

#define BS    32
#define LQ    300
#define NH    8
#define HD    32
#define NPTS  4
#define EMB   256
#define SUMP  16
#define NCOLS 384          // 256 offset cols + 128 attn cols
#define TOTAL 8500         // 80*80 + 40*40 + 20*20 + 10*10

// ---------------------------------------------------------------------------
// Kernel 1: P[9600 x 384] = query[9600 x 256] @ [W_off | W_attn] + [b_off|b_attn]
// f32 WMMA 16x16x4, K-loop of 64 steps. One block = 16(M) x 128(N) strip,
// 8 waves x (16x16 tile each). Query tile staged in padded LDS.
// ---------------------------------------------------------------------------
__global__ __launch_bounds__(256) void msda_gemm_kernel(
    const float* __restrict__ query,
    const float* __restrict__ W_off,  const float* __restrict__ b_off,
    const float* __restrict__ W_attn, const float* __restrict__ b_attn,
    float* __restrict__ P)
{
    __shared__ float qtile[16][EMB + 1];    // +1 pad: row stride 257 -> bank-conflict-free
    const int mtile  = blockIdx.x;          // 0..599  (9600/16)
    const int nstrip = blockIdx.y;          // 0..2    (384/128)
    const int tid    = threadIdx.x;
    const int row0   = mtile * 16;

    // cooperative load of the 16x256 query tile (coalesced)
    #pragma unroll
    for (int i = 0; i < 16; ++i) {
        const int idx = tid + i * 256;
        const int r = idx >> 8;
        const int c = idx & 255;
        qtile[r][c] = query[(row0 + r) * EMB + c];
    }
    __syncthreads();

    const int wave = tid >> 5;
    const int lane = tid & 31;
    const int n    = lane & 15;   // row (A) / col (B,C,D) within tile
    const int grp  = lane >> 4;   // K-pair select (A,B) / M+8 select (C,D)

    // select which weight matrix this strip reads
    const float* Wb;  const float* Bb;  int ldw, coloff;
    if (nstrip < 2) { Wb = W_off;  Bb = b_off;  ldw = 256; coloff = nstrip * 128; }
    else            { Wb = W_attn; Bb = b_attn; ldw = 128; coloff = 0; }
    const int colw = coloff + wave * 16 + n;   // column within selected W / bias

    v8f acc = {};
    for (int k = 0; k < EMB; k += 4) {
        // A 16x4 f32 fragment (ISA 7.12.2): v0 = A[M=n][k+2*grp], v1 = A[M=n][k+2*grp+1]
        v2f a, bfrag;
        a.x = qtile[n][k + 2 * grp];
        a.y = qtile[n][k + 2 * grp + 1];
        // B 4x16 f32 fragment: v0 = B[K=k+2*grp][N=n], v1 = B[K=k+2*grp+1][N=n]
        bfrag.x = Wb[(k + 2 * grp)     * ldw + colw];
        bfrag.y = Wb[(k + 2 * grp + 1) * ldw + colw];
        acc = __builtin_amdgcn_wmma_f32_16x16x4_f32(
                  /*neg_a=*/false, a, /*neg_b=*/false, bfrag,
                  /*c_mod=*/(short)0, acc, /*reuse_a=*/false, /*reuse_b=*/false);
    }

    const float bias = Bb[colw];
    const int   colg = nstrip * 128 + wave * 16 + n;
    // D layout: VGPR v holds M = v + 8*grp, N = n
    #pragma unroll
    for (int v = 0; v < 8; ++v) {
        const int r = row0 + v + 8 * grp;
        P[r * NCOLS + colg] = acc[v] + bias;
    }
}

// ---------------------------------------------------------------------------
// Kernel 2: softmax over 16 points per head + bilinear gather + weighted sum.
// Block = one (b,q): 8 waves, wave = head, lane = channel d.
// value[b,h,d,:] is spatially contiguous per lane -> corner pairs adjacent.
// ---------------------------------------------------------------------------
__global__ __launch_bounds__(256) void msda_sample_kernel(
    const float* __restrict__ P,
    const float* __restrict__ refp,     // [BS, LQ, 1, 4]
    const float* __restrict__ value,    // [BS, NH, HD, TOTAL]
    float* __restrict__ out)            // [BS, LQ, NH*HD]
{
    const int q   = blockIdx.x;
    const int b   = blockIdx.y;
    const int tid = threadIdx.x;
    const int h   = tid >> 5;
    const int d   = tid & 31;
    const int row = b * LQ + q;

    const float* Prow = P + (size_t)row * NCOLS;
    const float4 rp = ((const float4*)refp)[row];   // (x, y, w, h)

    // softmax over this head's 16 logits (broadcast loads, redundant per lane)
    float lg[SUMP];
    float mx = -1e30f;
    #pragma unroll
    for (int p = 0; p < SUMP; ++p) {
        lg[p] = Prow[EMB + h * SUMP + p];
        mx = fmaxf(mx, lg[p]);
    }
    float s = 0.0f;
    #pragma unroll
    for (int p = 0; p < SUMP; ++p) { lg[p] = __expf(lg[p] - mx); s += lg[p]; }
    const float inv = 1.0f / s;

    const float* vbase =
        value + (size_t)((b * NH + h) * HD + d) * TOTAL;

    const int Wl[4]  = {80, 40, 20, 10};         // square levels
    const int SPo[4] = {0, 6400, 8000, 8400};    // level start offsets

    float acc = 0.0f;
    #pragma unroll
    for (int lvl = 0; lvl < 4; ++lvl) {
        const int Wd = Wl[lvl];
        const int Hd = Wl[lvl];
        const float* vlev = vbase + SPo[lvl];
        const float fW = (float)Wd, fH = (float)Hd;
        #pragma unroll
        for (int pp = 0; pp < NPTS; ++pp) {
            const int p = lvl * NPTS + pp;
            const float ox = Prow[(h * SUMP + p) * 2 + 0];
            const float oy = Prow[(h * SUMP + p) * 2 + 1];
            // loc = ref_xy + off * (1/NPTS) * ref_wh * OFFSET_SCALE ; x = loc*W - 0.5
            const float lx = rp.x + ox * 0.125f * rp.z;
            const float ly = rp.y + oy * 0.125f * rp.w;
            const float x = lx * fW - 0.5f;
            const float y = ly * fH - 0.5f;
            const float xf = floorf(x), yf = floorf(y);
            const int   x0 = (int)xf,  y0 = (int)yf;
            const float fx = x - xf,   fy = y - yf;
            const int   x1 = x0 + 1,   y1 = y0 + 1;
            const bool vx0 = (x0 >= 0) && (x0 < Wd);
            const bool vx1 = (x1 >= 0) && (x1 < Wd);
            const bool vy0 = (y0 >= 0) && (y0 < Hd);
            const bool vy1 = (y1 >= 0) && (y1 < Hd);
            const float v00 = (vx0 && vy0) ? vlev[y0 * Wd + x0] : 0.0f;
            const float v10 = (vx1 && vy0) ? vlev[y0 * Wd + x1] : 0.0f;
            const float v01 = (vx0 && vy1) ? vlev[y1 * Wd + x0] : 0.0f;
            const float v11 = (vx1 && vy1) ? vlev[y1 * Wd + x1] : 0.0f;
            const float sample = (1.0f - fx) * (1.0f - fy) * v00
                               +         fx  * (1.0f - fy) * v10
                               + (1.0f - fx) *         fy  * v01
                               +         fx  *         fy  * v11;
            acc = fmaf(lg[p] * inv, sample, acc);
        }
    }
    out[(size_t)row * (NH * HD) + h * HD + d] = acc;
}

// ---------------------------------------------------------------------------
extern "C" void kernel_launch(void* const* d_in, const int* in_sizes, int n_in,
                              void* d_out, int out_size, void* d_ws, size_t ws_size,
                              hipStream_t stream) {
    const float* query  = (const float*)d_in[0];
    const float* refp   = (const float*)d_in[1];
    const float* value  = (const float*)d_in[2];
    // d_in[3] = value_spatial_shapes (int32) -- compile-time constants, unused
    const float* W_off  = (const float*)d_in[4];
    const float* b_off  = (const float*)d_in[5];
    const float* W_attn = (const float*)d_in[6];
    const float* b_attn = (const float*)d_in[7];
    float* out = (float*)d_out;
    float* P   = (float*)d_ws;   // 9600 x 384 floats = 14.75 MB scratch

    dim3 g1((BS * LQ) / 16, NCOLS / 128);   // (600, 3)
    msda_gemm_kernel<<<g1, dim3(256), 0, stream>>>(query, W_off, b_off,
                                                   W_attn, b_attn, P);

    dim3 g2(LQ, BS);                         // (300, 32)
    msda_sample_kernel<<<g2, dim3(256), 0, stream>>>(P, refp, value, out);
}